// ComplEx_8924942041805
// MI455X (gfx1250) — compile-verified
//
#include <hip/hip_runtime.h>

typedef __attribute__((ext_vector_type(2))) float v2f;
typedef __attribute__((ext_vector_type(4))) float v4f;
typedef __attribute__((ext_vector_type(8))) float v8f;

#define EMB   512      // K
#define HALF  256
#define MB    1024     // GEMM M (batch)
#define NE    200000   // GEMM N (entities)

#define BM    256      // workgroup M tile
#define BN    64       // workgroup N tile
#define KC    32       // K chunk staged in LDS
#define KCP   36       // padded LDS row stride (floats): conflict-free b64 frag reads

// ---------------------------------------------------------------------------
// Kernel 1: A[b, 0:256] = re_h*re_r - im_h*im_r ; A[b, 256:512] = re_h*im_r + im_h*re_r
// ---------------------------------------------------------------------------
__global__ __launch_bounds__(256) void complex_prep(
    const int* __restrict__ triples, const float* __restrict__ ent,
    const float* __restrict__ rel, float* __restrict__ A) {
  const int b = blockIdx.x;      // 0..1023
  const int j = threadIdx.x;     // 0..255
  const int h = triples[b * 3 + 0];
  const int r = triples[b * 3 + 1];
  const float* hp = ent + (size_t)h * EMB;
  const float* rp = rel + (size_t)r * EMB;
  const float re_h = hp[j];
  const float im_h = hp[j + HALF];
  const float re_r = rp[j];
  const float im_r = rp[j + HALF];
  A[(size_t)b * EMB + j]        = re_h * re_r - im_h * im_r;
  A[(size_t)b * EMB + j + HALF] = re_h * im_r + im_h * re_r;
}

// ---------------------------------------------------------------------------
// Kernel 2: C[1024 x 200000] = A[1024 x 512] * E[200000 x 512]^T  (f32 WMMA)
// 8 waves/WG; wave w owns rows [w*32, w*32+32) x all 64 cols of the tile:
// 2x4 grid of 16x16 accumulators. K advanced 4 at a time via wmma_f32_16x16x4_f32.
// ---------------------------------------------------------------------------
__global__ __launch_bounds__(256) void complex_score_gemm(
    const float* __restrict__ A, const float* __restrict__ E,
    float* __restrict__ C) {
  __shared__ float ldsA[BM * KCP];   // 36 KiB
  __shared__ float ldsB[BN * KCP];   //  9 KiB

  const int tid   = threadIdx.x;
  const int lane  = tid & 31;
  const int wave  = tid >> 5;          // 0..7
  const int lrow  = lane & 15;         // row/col within fragment
  const int khalf = (lane >> 4) * 2;   // 0: K={4k,4k+1}  2: K={4k+2,4k+3}
  const int m0 = blockIdx.y * BM;
  const int n0 = blockIdx.x * BN;

  // Global staging map: element e = tid + 256*i -> row = e>>3, col4 = e&7
  // => one b128 per lane; 32 lanes cover 4 full rows (coalesced 128B lines).
  const int grow = tid >> 3;           // 0..31
  const int gcol = (tid & 7) * 4;      // 0,4,...,28

  const float* Ag = A + ((size_t)(m0 + grow)) * EMB + gcol;
  const float* Eg = E + ((size_t)(n0 + grow)) * EMB + gcol;

  v4f aR[8];   // 256 rows / 32 = 8 b128 per thread per chunk
  v4f bR[2];   //  64 rows / 32 = 2 b128 per thread per chunk

  v8f acc[2][4];
#pragma unroll
  for (int mt = 0; mt < 2; ++mt)
#pragma unroll
    for (int nt = 0; nt < 4; ++nt)
      acc[mt][nt] = {};

  // preload chunk 0 into registers
#pragma unroll
  for (int i = 0; i < 8; ++i) aR[i] = *(const v4f*)(Ag + (size_t)(32 * i) * EMB);
#pragma unroll
  for (int i = 0; i < 2; ++i) bR[i] = *(const v4f*)(Eg + (size_t)(32 * i) * EMB);

  const int NCH = EMB / KC;  // 16
  for (int kc = 0; kc < NCH; ++kc) {
    // stage current chunk into LDS
#pragma unroll
    for (int i = 0; i < 8; ++i)
      *(v4f*)(&ldsA[(grow + 32 * i) * KCP + gcol]) = aR[i];
#pragma unroll
    for (int i = 0; i < 2; ++i)
      *(v4f*)(&ldsB[(grow + 32 * i) * KCP + gcol]) = bR[i];
    __syncthreads();

    // prefetch next chunk (overlaps with WMMA work below)
    if (kc + 1 < NCH) {
      const float* Agn = Ag + (kc + 1) * KC;
      const float* Egn = Eg + (kc + 1) * KC;
#pragma unroll
      for (int i = 0; i < 8; ++i) aR[i] = *(const v4f*)(Agn + (size_t)(32 * i) * EMB);
#pragma unroll
      for (int i = 0; i < 2; ++i) bR[i] = *(const v4f*)(Egn + (size_t)(32 * i) * EMB);
    }

    // compute on staged chunk: 8 K-steps x 8 accumulator tiles = 64 WMMAs
#pragma unroll
    for (int ks = 0; ks < KC / 4; ++ks) {
      v2f af[2], bf[4];
#pragma unroll
      for (int mt = 0; mt < 2; ++mt)
        af[mt] = *(const v2f*)(&ldsA[(wave * 32 + mt * 16 + lrow) * KCP + ks * 4 + khalf]);
#pragma unroll
      for (int nt = 0; nt < 4; ++nt)
        bf[nt] = *(const v2f*)(&ldsB[(nt * 16 + lrow) * KCP + ks * 4 + khalf]);
#pragma unroll
      for (int mt = 0; mt < 2; ++mt)
#pragma unroll
        for (int nt = 0; nt < 4; ++nt)
          acc[mt][nt] = __builtin_amdgcn_wmma_f32_16x16x4_f32(
              /*neg_a=*/false, af[mt], /*neg_b=*/false, bf[nt],
              /*c_mod=*/(short)0, acc[mt][nt],
              /*reuse_a=*/false, /*reuse_b=*/false);
    }
    __syncthreads();
  }

  // write out: C-frag VGPR j, lanes 0-15 -> (M=j, N=lane); lanes 16-31 -> (M=j+8, N=lane-16)
  float* Cw = C + (size_t)(m0 + wave * 32) * NE + n0;
#pragma unroll
  for (int mt = 0; mt < 2; ++mt) {
#pragma unroll
    for (int nt = 0; nt < 4; ++nt) {
      float* cp = Cw + (size_t)(mt * 16 + (lane >> 4) * 8) * NE + nt * 16 + lrow;
#pragma unroll
      for (int j = 0; j < 8; ++j)
        cp[(size_t)j * NE] = acc[mt][nt][j];
    }
  }
}

// ---------------------------------------------------------------------------
extern "C" void kernel_launch(void* const* d_in, const int* in_sizes, int n_in,
                              void* d_out, int out_size, void* d_ws, size_t ws_size,
                              hipStream_t stream) {
  const int*   triples = (const int*)d_in[0];    // [1024][3] int
  const float* ent     = (const float*)d_in[1];  // [200000][512] f32
  const float* rel     = (const float*)d_in[2];  // [1000][512] f32
  float* A = (float*)d_ws;    // [1024][512] f32 staging (2 MB)
  float* C = (float*)d_out;   // [1024][200000] f32

  complex_prep<<<dim3(MB), dim3(256), 0, stream>>>(triples, ent, rel, A);

  dim3 grid(NE / BN /*3125*/, MB / BM /*4*/);
  complex_score_gemm<<<grid, dim3(256), 0, stream>>>(A, ent, C);
}